// ProposalLayer_77910706749627
// MI455X (gfx1250) — compile-verified
//
#include <hip/hip_runtime.h>
#include <stdint.h>

#define N_PRE     12000
#define N_POST    2000
#define NMS_T     0.7f
#define MIN_SZ    16.0f
#define NBLK      375        // N_PRE / 32 exactly

typedef unsigned long long u64;
typedef unsigned int       u32;

// ---------- gfx1250 async global->LDS path (probe-verified on this toolchain) ------
#if defined(__has_builtin)
#if __has_builtin(__builtin_amdgcn_global_load_async_to_lds_b128) && \
    __has_builtin(__builtin_amdgcn_s_wait_asynccnt)
#define USE_ASYNC_LDS 1
#endif
#endif
#ifndef USE_ASYNC_LDS
#define USE_ASYNC_LDS 0
#endif

#if defined(__HIP_DEVICE_COMPILE__) && USE_ASYNC_LDS
// builtin signature (from clang diagnostic): param0 = int4 AS(1)*, param1 = int4 AS(3)*
typedef int v4i_t __attribute__((vector_size(16)));
typedef __attribute__((address_space(1))) v4i_t g_v4i;
typedef __attribute__((address_space(3))) v4i_t l_v4i;
// generic LDS addr: low 32 bits are the LDS byte offset (ISA 10.2 aperture rules)
__device__ static inline g_v4i* to_gbl(const void* p) {
  return (g_v4i*)(uintptr_t)p;
}
__device__ static inline l_v4i* to_lds(void* p) {
  return (l_v4i*)(u32)(uintptr_t)p;
}
#endif

// ---------------- Kernel 1: decode + clip + min-size filter + sort-key pack --------
__global__ void decode_kernel(const float4* __restrict__ anchors,
                              const float4* __restrict__ offs,
                              const float*  __restrict__ scores,
                              const int* pH, const int* pW, const int* pS,
                              float4* __restrict__ roi,
                              u64*    __restrict__ keys,
                              int n, int padn) {
  int i = blockIdx.x * blockDim.x + threadIdx.x;
  if (i >= padn) return;
  if (i >= n) { keys[i] = 0ull; return; }   // padding sorts to the bottom

  float H = (float)*pH, W = (float)*pW;
  float ms = MIN_SZ * (float)*pS;

  float4 a = anchors[i], o = offs[i];
  float ah  = a.z - a.x,        aw  = a.w - a.y;
  float acy = a.x + 0.5f * ah,  acx = a.y + 0.5f * aw;
  float cy  = o.x * ah + acy,   cx  = o.y * aw + acx;
  float h   = __expf(o.z) * ah, w_  = __expf(o.w) * aw;

  float y1 = fminf(fmaxf(cy - 0.5f * h,  0.0f), H);
  float x1 = fminf(fmaxf(cx - 0.5f * w_, 0.0f), W);
  float y2 = fminf(fmaxf(cy + 0.5f * h,  0.0f), H);
  float x2 = fminf(fmaxf(cx + 0.5f * w_, 0.0f), W);
  roi[i] = make_float4(y1, x1, y2, x2);

  bool valid = ((y2 - y1) >= ms) && ((x2 - x1) >= ms);
  float s = valid ? scores[i] : -INFINITY;
  u32 b = __float_as_uint(s);
  u32 ord = (b & 0x80000000u) ? ~b : (b | 0x80000000u);  // monotone float->uint
  // descending score; ties -> smaller index first (matches lax.top_k)
  keys[i] = ((u64)ord << 32) | (u64)(0xFFFFFFFFu - (u32)i);
}

// ---------------- Kernel 2a: bitonic global pass (j >= 512, descending) ------------
__global__ void bitonic_kernel(u64* __restrict__ keys, int k, int j, int padn) {
  int tid = blockIdx.x * blockDim.x + threadIdx.x;
  if (tid >= padn) return;
  int ixj = tid ^ j;
  if (ixj <= tid) return;
  u64 x = keys[tid], y = keys[ixj];
  bool up = (tid & k) == 0;
  bool sw = up ? (x < y) : (x > y);       // descending overall
  if (sw) { keys[tid] = y; keys[ixj] = x; }
}

// ---------------- Kernel 2b: bitonic fused LDS passes (j = jstart..1) --------------
// 512-key tile (4KB LDS) per 256-thread block; fuses the 9 innermost stages.
__global__ void __launch_bounds__(256)
bitonic_fused_kernel(u64* __restrict__ keys, int k, int jstart) {
  __shared__ u64 tile[512];
  int base = blockIdx.x * 512;
  int t = threadIdx.x;
  tile[t]       = keys[base + t];
  tile[t + 256] = keys[base + t + 256];
  __syncthreads();
  for (int j = jstart; j >= 1; j >>= 1) {
    int p = ((t & ~(j - 1)) << 1) | (t & (j - 1));   // bit j of p is clear
    int q = p | j;
    bool up = ((base + p) & k) == 0;                 // base|p since p < 512
    u64 x = tile[p], y = tile[q];
    bool sw = up ? (x < y) : (x > y);                // descending
    if (sw) { tile[p] = y; tile[q] = x; }
    __syncthreads();
  }
  keys[base + t]       = tile[t];
  keys[base + t + 256] = tile[t + 256];
}

// ---------------- Kernel 3: gather top-N_PRE boxes + keep0 ballot (wave32) ---------
__global__ void gather_kernel(const u64*    __restrict__ keys,
                              const float4* __restrict__ roi,
                              float4* __restrict__ top,
                              u32*    __restrict__ keep0, int n) {
  int r = blockIdx.x * blockDim.x + threadIdx.x;
  if (r >= N_PRE) return;
  u64 key = keys[r];
  u32 idx = 0xFFFFFFFFu - (u32)(key & 0xFFFFFFFFull);
  if (idx >= (u32)n) idx = 0;             // safety (padding should never reach here)
  top[r] = roi[idx];
  u32 ordv = (u32)(key >> 32);
  bool valid = ordv > 0x007FFFFFu;        // strictly > ordered(-inf)
  unsigned long long bal = __ballot(valid);
  if ((threadIdx.x & 31) == 0) keep0[r >> 5] = (u32)bal;
}

// ---------------- Kernel 4: NMS suppression bitmask, 256 rows x 32 cols per block --
// One async-DMA'd 512B column tile in LDS is reused by all 8 waves of the block.
__global__ void __launch_bounds__(256)
nms_mask_kernel(const float4* __restrict__ boxes, u32* __restrict__ mask) {
  int cxb = blockIdx.x;                   // column block: cols [cxb*32, cxb*32+31]
  int r0  = blockIdx.y * 256;             // first row of this block
  int t   = threadIdx.x;
  int r   = r0 + t;

  // entire block strictly below the diagonal (max col < min row): no j>i bits
  if (cxb * 32 + 31 < r0) {
    if (r < N_PRE) mask[(size_t)r * NBLK + cxb] = 0u;
    return;
  }

  __shared__ float4 ct[32];
  float4 rb = make_float4(0.f, 0.f, 0.f, 0.f);
#if USE_ASYNC_LDS
  if (t < 32)                             // wave 0 (full EXEC) issues the async DMA
    __builtin_amdgcn_global_load_async_to_lds_b128(
        to_gbl(&boxes[cxb * 32 + t]), to_lds(&ct[t]), 0, 0);
  if (r < N_PRE) rb = boxes[r];           // overlap row load with async copy
  __builtin_amdgcn_s_wait_asynccnt(0);
  __syncthreads();
#else
  if (t < 32) ct[t] = boxes[cxb * 32 + t];
  if (r < N_PRE) rb = boxes[r];
  __syncthreads();
#endif
  if (r >= N_PRE) return;

  u32 m = 0;
  float rArea = (rb.z - rb.x) * (rb.w - rb.y);
#pragma unroll 8
  for (int jj = 0; jj < 32; ++jj) {
    int c = cxb * 32 + jj;
    if (c <= r) continue;                 // only j > i suppresses
    float4 cb = ct[jj];
    float cArea = (cb.z - cb.x) * (cb.w - cb.y);
    float ty = fmaxf(rb.x, cb.x), tx = fmaxf(rb.y, cb.y);
    float by = fminf(rb.z, cb.z), bx = fminf(rb.w, cb.w);
    float inter = fmaxf(by - ty, 0.0f) * fmaxf(bx - tx, 0.0f);
    float uni = fmaxf(rArea + cArea - inter, 1e-9f);
    if (inter > NMS_T * uni) m |= (1u << jj);
  }
  mask[(size_t)r * NBLK + cxb] = m;
}

// ---------------- Kernel 5: sequential greedy scan (single wave) -------------------
__global__ void __launch_bounds__(32)
nms_scan_kernel(const float4* __restrict__ top,
                const u32*    __restrict__ keep0,
                const u32*    __restrict__ mask,
                float* __restrict__ out) {
  __shared__ u32 remv[NBLK];
  int lane = threadIdx.x;
  for (int t = lane; t < NBLK; t += 32) remv[t] = 0u;
  for (int t = lane; t < N_POST * 4; t += 32) out[t] = 0.0f;  // zero-pad output
  __syncthreads();

  const float* tb = (const float*)top;
  int rank = 0;
  for (int i = 0; i < N_PRE && rank < N_POST; ++i) {
    if (i + 1 < N_PRE)
      __builtin_prefetch(&mask[(size_t)(i + 1) * NBLK], 0, 0);  // global_prefetch_b8
    u32 rm = remv[i >> 5];
    u32 k0 = keep0[i >> 5];
    bool kept = ((k0 >> (i & 31)) & 1u) && !((rm >> (i & 31)) & 1u);
    if (kept) {
      if (lane < 4) out[rank * 4 + lane] = tb[i * 4 + lane];
      ++rank;
      const u32* row = &mask[(size_t)i * NBLK];
      for (int t = lane; t < NBLK; t += 32) remv[t] |= row[t];
    }
    __syncthreads();
  }
}

// ----------------------------------- launcher --------------------------------------
extern "C" void kernel_launch(void* const* d_in, const int* in_sizes, int n_in,
                              void* d_out, int out_size, void* d_ws, size_t ws_size,
                              hipStream_t stream) {
  const float4* anchors = (const float4*)d_in[0];
  const float4* offs    = (const float4*)d_in[1];
  const float*  scores  = (const float*)d_in[2];
  const int*    pH      = (const int*)d_in[3];
  const int*    pW      = (const int*)d_in[4];
  const int*    pS      = (const int*)d_in[5];
  float*        out     = (float*)d_out;

  int n = in_sizes[0] / 4;                 // anchors are (N,4)
  int padn = 1; while (padn < n) padn <<= 1;

  char* w = (char*)d_ws;
  size_t off = 0;
  auto carve = [&](size_t bytes) -> void* {
    void* p = w + off;
    off = (off + bytes + 255) & ~(size_t)255;
    return p;
  };
  float4* roi   = (float4*)carve((size_t)n * 16);
  u64*    keys  = (u64*)   carve((size_t)padn * 8);
  float4* top   = (float4*)carve((size_t)N_PRE * 16);
  u32*    keep0 = (u32*)   carve((size_t)NBLK * 4);
  u32*    mask  = (u32*)   carve((size_t)N_PRE * NBLK * 4);
  (void)ws_size; (void)n_in; (void)out_size;

  const int TB = 256;
  int gpad = (padn + TB - 1) / TB;
  decode_kernel<<<gpad, TB, 0, stream>>>(anchors, offs, scores, pH, pW, pS,
                                         roi, keys, n, padn);

  // hybrid bitonic sort: global passes for j>=512, LDS-fused passes for j<=256
  if (padn >= 512) {
    int gfuse = padn / 512;
    for (int k = 2; k <= padn; k <<= 1) {
      for (int j = k >> 1; j >= 512; j >>= 1)
        bitonic_kernel<<<gpad, TB, 0, stream>>>(keys, k, j, padn);
      int jstart = (k >> 1 < 256) ? (k >> 1) : 256;
      bitonic_fused_kernel<<<gfuse, 256, 0, stream>>>(keys, k, jstart);
    }
  } else {
    for (int k = 2; k <= padn; k <<= 1)
      for (int j = k >> 1; j >= 1; j >>= 1)
        bitonic_kernel<<<gpad, TB, 0, stream>>>(keys, k, j, padn);
  }

  gather_kernel<<<(N_PRE + TB - 1) / TB, TB, 0, stream>>>(keys, roi, top, keep0, n);

  dim3 gm((unsigned)NBLK, (unsigned)((N_PRE + 255) / 256));
  nms_mask_kernel<<<gm, 256, 0, stream>>>(top, mask);

  nms_scan_kernel<<<1, 32, 0, stream>>>(top, keep0, mask, out);
}